// GraphConvolution1_11836929868275
// MI455X (gfx1250) — compile-verified
//
#include <hip/hip_runtime.h>
#include <hip/hip_bf16.h>

#define NNODES 50000
#define NEDGES 800000
#define IN_DIM 256
#define OUT_DIM 128
#define LEAKY 0.2f

typedef __attribute__((ext_vector_type(2))) float v2f;
typedef __attribute__((ext_vector_type(8))) float v8f;

// ---- order-preserving float <-> uint for atomicMax-based segment max ----
__device__ __forceinline__ unsigned f2ord(float f) {
    unsigned u = __float_as_uint(f);
    return (u & 0x80000000u) ? ~u : (u | 0x80000000u);
}
__device__ __forceinline__ float ord2f(unsigned u) {
    return (u & 0x80000000u) ? __uint_as_float(u & 0x7FFFFFFFu) : __uint_as_float(~u);
}

// ============================================================================
// K1: xt0 = x @ W + bias  via V_WMMA_F32_16X16X4_F32 (full fp32 precision)
// Block = 128 threads = 4 waves; each wave owns one 16x16 D tile.
// A-row index is CLAMPED (not predicated) so the inner loop has no EXEC
// manipulation and loads can be pipelined ahead of the WMMAs. Two
// accumulators break the WMMA D->C dependency chain.
// ============================================================================
__global__ __launch_bounds__(128) void gemm_xw_wmma(
    const float* __restrict__ x, const float* __restrict__ W,
    const float* __restrict__ bias, float* __restrict__ xt0)
{
    __shared__ float Bs[IN_DIM * 16]; // Bs[(k>>1)*32 + n*2 + (k&1)] = W[k][ncol0+n]
    const int t = threadIdx.x;
    const int ncol0 = blockIdx.y * 16;

    for (int idx = t; idx < IN_DIM * 16; idx += 128) {
        int k = idx >> 4;
        int n = idx & 15;
        Bs[(k >> 1) * 32 + n * 2 + (k & 1)] = W[k * OUT_DIM + ncol0 + n];
    }
    __syncthreads();

    const int wave  = t >> 5;
    const int lane  = t & 31;
    const int m0    = blockIdx.x * 64 + wave * 16;
    const int nloc  = lane & 15;
    const int khalf = (lane >> 4) * 2;              // K-pair select per lane half
    int arow = m0 + nloc;                           // A: row = lane&15, both halves
    if (arow >= NNODES) arow = NNODES - 1;          // clamp: loads always in-bounds
    const float* __restrict__ xrow = x + (size_t)arow * IN_DIM + khalf;
    const float* __restrict__ brow = &Bs[(khalf >> 1) * 32 + nloc * 2];

    v8f c0 = {};
    v8f c1 = {};
    #pragma unroll 8
    for (int k0 = 0; k0 < IN_DIM; k0 += 8) {
        v2f a0 = *(const v2f*)(xrow + k0);
        v2f b0 = *(const v2f*)(brow + k0 * 8);      // (k0>>2)*32 floats
        v2f a1 = *(const v2f*)(xrow + k0 + 4);
        v2f b1 = *(const v2f*)(brow + (k0 + 4) * 8);
        // (neg_a, A, neg_b, B, c_mod, C, reuse_a, reuse_b)
        c0 = __builtin_amdgcn_wmma_f32_16x16x4_f32(false, a0, false, b0,
                                                   (short)0, c0, false, false);
        c1 = __builtin_amdgcn_wmma_f32_16x16x4_f32(false, a1, false, b1,
                                                   (short)0, c1, false, false);
    }

    const int ncol = ncol0 + nloc;
    const float bn = bias[ncol];
    #pragma unroll
    for (int v = 0; v < 8; ++v) {
        int m = m0 + v + ((lane >> 4) << 3);        // D: m = v + 8*(lane>=16)
        if (m < NNODES)
            xt0[(size_t)m * OUT_DIM + ncol] = c0[v] + c1[v] + bn;
    }
}

// ============================================================================
// K2: zero xt / neighbor_in, init softmax state, compute per-node gate
// ============================================================================
__global__ void init_state(float* __restrict__ xt, float* __restrict__ nin,
                           float* __restrict__ gate,
                           float* __restrict__ sum_out, float* __restrict__ sum_in,
                           unsigned* __restrict__ max_out, unsigned* __restrict__ max_in,
                           const float* __restrict__ outdeg, const float* __restrict__ indeg,
                           const float* __restrict__ wdo, const float* __restrict__ wdi)
{
    const int stride = gridDim.x * blockDim.x;
    const int i0 = blockIdx.x * blockDim.x + threadIdx.x;
    const float a = wdo[0], b = wdi[0];
    for (int j = i0; j < NNODES * OUT_DIM; j += stride) { xt[j] = 0.f; nin[j] = 0.f; }
    for (int j = i0; j < NNODES; j += stride) {
        float g = a * outdeg[j] + b * indeg[j];
        gate[j] = 1.f / (1.f + expf(-g));
        sum_out[j] = 0.f; sum_in[j] = 0.f;
        max_out[j] = 0u;  max_in[j] = 0u;   // 0 < any finite encoding
    }
}

// ============================================================================
// K3: xt[snd] += adj * xt0[rcv]  (wave per edge, float4 per lane, f32 atomics)
// ============================================================================
__global__ __launch_bounds__(128) void scatter_agg(
    const int* __restrict__ edges, const float* __restrict__ adj_vals,
    const float* __restrict__ xt0, float* __restrict__ xt)
{
    const int e = blockIdx.x * 4 + (threadIdx.x >> 5);
    if (e >= NEDGES) return;
    const int lane = threadIdx.x & 31;
    const int snd = edges[2 * e + 0];
    const int rcv = edges[2 * e + 1];
    const float v = adj_vals[e];
    float4 d = ((const float4*)(xt0 + (size_t)rcv * OUT_DIM))[lane];
    float* dst = xt + (size_t)snd * OUT_DIM + lane * 4;
    atomicAdd(dst + 0, v * d.x);
    atomicAdd(dst + 1, v * d.y);
    atomicAdd(dst + 2, v * d.z);
    atomicAdd(dst + 3, v * d.w);
}

// ============================================================================
// K4: per-node attention dots  s_a = xt . phi[0:128], s_b = xt . phi[128:256]
// ============================================================================
__global__ __launch_bounds__(128) void node_dots(
    const float* __restrict__ xt, const float* __restrict__ phi,
    float* __restrict__ s_a, float* __restrict__ s_b)
{
    const int node = blockIdx.x * 4 + (threadIdx.x >> 5);
    if (node >= NNODES) return;
    const int lane = threadIdx.x & 31;
    float4 r  = ((const float4*)(xt + (size_t)node * OUT_DIM))[lane];
    float4 pa = ((const float4*)phi)[lane];
    float4 pb = ((const float4*)(phi + OUT_DIM))[lane];
    float sa = r.x * pa.x + r.y * pa.y + r.z * pa.z + r.w * pa.w;
    float sb = r.x * pb.x + r.y * pb.y + r.z * pb.z + r.w * pb.w;
    #pragma unroll
    for (int o = 16; o > 0; o >>= 1) {
        sa += __shfl_xor(sa, o, 32);
        sb += __shfl_xor(sb, o, 32);
    }
    if (lane == 0) { s_a[node] = sa; s_b[node] = sb; }
}

// ============================================================================
// K5: att[e] = leaky_relu(s_a[rcv] + s_b[snd] + b); segment max (ordered-uint)
// ============================================================================
__global__ void att_and_max(const int* __restrict__ edges,
                            const float* __restrict__ s_a, const float* __restrict__ s_b,
                            const float* __restrict__ bias_att,
                            float* __restrict__ att,
                            unsigned* __restrict__ max_out, unsigned* __restrict__ max_in)
{
    const int e = blockIdx.x * blockDim.x + threadIdx.x;
    if (e >= NEDGES) return;
    const int snd = edges[2 * e + 0];
    const int rcv = edges[2 * e + 1];
    float v = s_a[rcv] + s_b[snd] + bias_att[0];
    v = (v > 0.f) ? v : LEAKY * v;
    att[e] = v;
    unsigned enc = f2ord(v);
    atomicMax(max_out + snd, enc);
    atomicMax(max_in + rcv, enc);
}

// ============================================================================
// K6: segment exp-sums
// ============================================================================
__global__ void exp_sums(const int* __restrict__ edges, const float* __restrict__ att,
                         const unsigned* __restrict__ max_out, const unsigned* __restrict__ max_in,
                         float* __restrict__ sum_out, float* __restrict__ sum_in)
{
    const int e = blockIdx.x * blockDim.x + threadIdx.x;
    if (e >= NEDGES) return;
    const int snd = edges[2 * e + 0];
    const int rcv = edges[2 * e + 1];
    const float v = att[e];
    atomicAdd(sum_out + snd, expf(v - ord2f(max_out[snd])));
    atomicAdd(sum_in  + rcv, expf(v - ord2f(max_in[rcv])));
}

__global__ void zero_buf(float* __restrict__ p, int n) {
    const int stride = gridDim.x * blockDim.x;
    for (int j = blockIdx.x * blockDim.x + threadIdx.x; j < n; j += stride) p[j] = 0.f;
}

// ============================================================================
// K7: neighbor_out[snd] += a_out * gate[rcv] * xt[rcv]
//     neighbor_in [rcv] += a_in  * gate[snd] * xt[snd]   (wave per edge)
// ============================================================================
__global__ __launch_bounds__(128) void scatter_neighbors(
    const int* __restrict__ edges, const float* __restrict__ att,
    const unsigned* __restrict__ max_out, const unsigned* __restrict__ max_in,
    const float* __restrict__ sum_out, const float* __restrict__ sum_in,
    const float* __restrict__ gate, const float* __restrict__ xt,
    float* __restrict__ nout, float* __restrict__ nin)
{
    const int e = blockIdx.x * 4 + (threadIdx.x >> 5);
    if (e >= NEDGES) return;
    const int lane = threadIdx.x & 31;
    const int snd = edges[2 * e + 0];
    const int rcv = edges[2 * e + 1];
    const float v = att[e];
    const float a_out = expf(v - ord2f(max_out[snd])) / sum_out[snd];
    const float a_in  = expf(v - ord2f(max_in[rcv]))  / sum_in[rcv];
    const float co = a_out * gate[rcv];
    const float ci = a_in  * gate[snd];
    float4 xr = ((const float4*)(xt + (size_t)rcv * OUT_DIM))[lane];
    float4 xs = ((const float4*)(xt + (size_t)snd * OUT_DIM))[lane];
    float* po = nout + (size_t)snd * OUT_DIM + lane * 4;
    float* pi = nin  + (size_t)rcv * OUT_DIM + lane * 4;
    atomicAdd(po + 0, co * xr.x); atomicAdd(po + 1, co * xr.y);
    atomicAdd(po + 2, co * xr.z); atomicAdd(po + 3, co * xr.w);
    atomicAdd(pi + 0, ci * xs.x); atomicAdd(pi + 1, ci * xs.y);
    atomicAdd(pi + 2, ci * xs.z); atomicAdd(pi + 3, ci * xs.w);
}

// ============================================================================
// K8: out = relu(w_self*xt + w_no*neighbor_out + w_ni*neighbor_in)
// ============================================================================
__global__ void finalize(const float* __restrict__ xt, const float* __restrict__ nout,
                         const float* __restrict__ nin,
                         const float* __restrict__ w_self, const float* __restrict__ w_no,
                         const float* __restrict__ w_ni, float* __restrict__ out)
{
    const int stride = gridDim.x * blockDim.x;
    const float a = w_self[0], b = w_no[0], c = w_ni[0];
    for (int j = blockIdx.x * blockDim.x + threadIdx.x; j < NNODES * OUT_DIM; j += stride) {
        float v = a * xt[j] + b * nout[j] + c * nin[j];
        out[j] = v > 0.f ? v : 0.f;
    }
}

// ============================================================================
extern "C" void kernel_launch(void* const* d_in, const int* in_sizes, int n_in,
                              void* d_out, int out_size, void* d_ws, size_t ws_size,
                              hipStream_t stream) {
    const float* x        = (const float*)d_in[0];
    const int*   edges    = (const int*)  d_in[1];
    const float* adj_vals = (const float*)d_in[2];
    const float* outdeg   = (const float*)d_in[3];
    const float* indeg    = (const float*)d_in[4];
    const float* W        = (const float*)d_in[5];
    const float* bias     = (const float*)d_in[6];
    const float* phi      = (const float*)d_in[7];
    const float* bias_att = (const float*)d_in[8];
    const float* wdo      = (const float*)d_in[9];
    const float* wdi      = (const float*)d_in[10];
    const float* w_self   = (const float*)d_in[11];
    const float* w_no     = (const float*)d_in[12];
    const float* w_ni     = (const float*)d_in[13];
    float* out = (float*)d_out;

    // workspace carving (floats)
    float* ws = (float*)d_ws;
    const size_t NM = (size_t)NNODES * OUT_DIM; // 6.4M
    float*    xt0     = ws;                 // reused as neighbor_out after K3
    float*    xt      = ws + NM;
    float*    nin     = ws + 2 * NM;
    float*    gate    = ws + 3 * NM;
    float*    s_a     = gate + NNODES;
    float*    s_b     = s_a + NNODES;
    float*    sum_out = s_b + NNODES;
    float*    sum_in  = sum_out + NNODES;
    unsigned* max_out = (unsigned*)(sum_in + NNODES);
    unsigned* max_in  = max_out + NNODES;
    float*    att     = (float*)(max_in + NNODES);
    float*    nout    = xt0;

    // K1: GEMM via WMMA f32
    gemm_xw_wmma<<<dim3((NNODES + 63) / 64, OUT_DIM / 16), 128, 0, stream>>>(x, W, bias, xt0);
    // K2: init
    init_state<<<1024, 256, 0, stream>>>(xt, nin, gate, sum_out, sum_in,
                                         max_out, max_in, outdeg, indeg, wdo, wdi);
    // K3: aggregate
    scatter_agg<<<(NEDGES + 3) / 4, 128, 0, stream>>>(edges, adj_vals, xt0, xt);
    // K4: per-node attention dots
    node_dots<<<(NNODES + 3) / 4, 128, 0, stream>>>(xt, phi, s_a, s_b);
    // K5: edge attention + segment max
    att_and_max<<<(NEDGES + 255) / 256, 256, 0, stream>>>(edges, s_a, s_b, bias_att,
                                                          att, max_out, max_in);
    // K6: exp sums
    exp_sums<<<(NEDGES + 255) / 256, 256, 0, stream>>>(edges, att, max_out, max_in,
                                                       sum_out, sum_in);
    // zero neighbor_out (reuses xt0 storage)
    zero_buf<<<1024, 256, 0, stream>>>(nout, NNODES * OUT_DIM);
    // K7: neighbor scatters
    scatter_neighbors<<<(NEDGES + 3) / 4, 128, 0, stream>>>(edges, att, max_out, max_in,
                                                            sum_out, sum_in, gate, xt,
                                                            nout, nin);
    // K8: finalize
    finalize<<<1024, 256, 0, stream>>>(xt, nout, nin, w_self, w_no, w_ni, out);
}